// SCN_11244224380965
// MI455X (gfx1250) — compile-verified
//
#include <hip/hip_runtime.h>
#include <hip/hip_bf16.h>

// ---------------------------------------------------------------------------
// out = (L @ x) @ W.T + b
//   Phase 1: zero agg (scratch, N*128 f32)
//   Phase 2: per-edge scatter: agg[rows[e]] += vals[e] * x[cols[e]]  (atomics)
//   Phase 3: WMMA fp32 GEMM: out = agg @ W.T + b  (v_wmma_f32_16x16x4_f32)
// ---------------------------------------------------------------------------

#define F_DIM 128

typedef float v2f __attribute__((ext_vector_type(2)));
typedef float v8f __attribute__((ext_vector_type(8)));

// -------------------------- Phase 1: zero scratch ---------------------------
__global__ __launch_bounds__(256) void SCN_zero_kernel(float4* __restrict__ p, int n4) {
    int i = blockIdx.x * blockDim.x + threadIdx.x;
    if (i < n4) p[i] = make_float4(0.f, 0.f, 0.f, 0.f);
}

// -------------------------- Phase 2: edge scatter ---------------------------
// One wave32 per edge: lane l handles features [4l, 4l+4). Edge scalars are
// forced uniform via readfirstlane so they become s_load (1 scalar load vs
// 32 redundant vector loads). Scatter uses native fp32 global atomic add.
__global__ __launch_bounds__(256) void SCN_scatter_kernel(
    const int* __restrict__ rows, const int* __restrict__ cols,
    const float* __restrict__ vals, const float* __restrict__ x,
    float* __restrict__ agg, int E) {
    int wid = __builtin_amdgcn_readfirstlane(blockIdx.x * 8 + (threadIdx.x >> 5));
    int lane = threadIdx.x & 31;
    if (wid >= E) return;                       // wave-uniform exit
    int   r = rows[wid];
    int   c = cols[wid];
    float v = vals[wid];
    const float4* xr = (const float4*)(x + (size_t)c * F_DIM);
    float4 m = xr[lane];
    float* dst = agg + (size_t)r * F_DIM + lane * 4;
    unsafeAtomicAdd(dst + 0, m.x * v);
    unsafeAtomicAdd(dst + 1, m.y * v);
    unsafeAtomicAdd(dst + 2, m.z * v);
    unsafeAtomicAdd(dst + 3, m.w * v);
}

// -------------------------- Phase 3: WMMA GEMM ------------------------------
// D(16x16) = A(16x4) x B(4x16) + C, fp32 exact path.
// A layout (ISA 16x4 f32): lane<16 -> row=lane,   holds K = k+0, k+1
//                          lane>=16 -> row=lane-16, holds K = k+2, k+3
// B layout (4x16 f32):     lane<16 -> col=lane,   holds K = k+0, k+1
//                          lane>=16 -> col=lane-16, holds K = k+2, k+3
// C/D layout:              vgpr i, lane<16 -> (M=i,   N=lane)
//                                  lane>=16 -> (M=i+8, N=lane-16)
// W is pre-packed into LDS in fragment order: Bfrag[f*32 + lane] is exactly
// lane's float2 for fragment f = kstep*8 + jtile  -> conflict-free ds_load_b64.
__global__ __launch_bounds__(256) void SCN_gemm_wmma_kernel(
    const float* __restrict__ agg, const float* __restrict__ W,
    const float* __restrict__ bias, float* __restrict__ out, int ntiles) {
    __shared__ v2f Bfrag[8192];                  // 64 KB, exactly full W

    const int t = threadIdx.x;
    // Cooperative fragment-pack of W: e = f*32 + L
    for (int e = t; e < 8192; e += 256) {
        int L     = e & 31;
        int f     = e >> 5;
        int kstep = f >> 3;
        int j     = f & 7;
        int row   = j * 16 + (L & 15);                    // out-feature o
        int col   = kstep * 4 + ((L >> 4) << 1);          // k offset
        v2f w2;
        w2.x = W[row * F_DIM + col];
        w2.y = W[row * F_DIM + col + 1];
        Bfrag[e] = w2;
    }
    __syncthreads();

    const int tile = blockIdx.x * 8 + (t >> 5);
    if (tile >= ntiles) return;                  // wave-uniform (EXEC stays all-1)
    const int lane  = t & 31;
    const int m     = lane & 15;
    const int khalf = lane >> 4;                 // 0 or 1 -> K offset 0 / 2
    const size_t nodeBase = (size_t)tile * 16;
    const float* Arow = agg + (nodeBase + m) * F_DIM + khalf * 2;

    v8f acc[8] = {};
    for (int kstep = 0; kstep < 32; ++kstep) {
        v2f a = *(const v2f*)(Arow + kstep * 4);
        const v2f* bp = &Bfrag[kstep * 256 + lane];
#pragma unroll
        for (int j = 0; j < 8; ++j) {
            acc[j] = __builtin_amdgcn_wmma_f32_16x16x4_f32(
                false, a, false, bp[j * 32], (short)0, acc[j], false, false);
        }
    }

#pragma unroll
    for (int j = 0; j < 8; ++j) {
        float bv = bias[j * 16 + m];
#pragma unroll
        for (int i = 0; i < 8; ++i) {
            out[(nodeBase + (size_t)(i + 8 * khalf)) * F_DIM + j * 16 + m] =
                acc[j][i] + bv;
        }
    }
}

// Scalar tail for N % 16 rows (N=100000 -> 0 rows, kept for generality)
__global__ __launch_bounds__(128) void SCN_gemm_tail_kernel(
    const float* __restrict__ agg, const float* __restrict__ W,
    const float* __restrict__ bias, float* __restrict__ out,
    int startRow, int N) {
    int idx = blockIdx.x * blockDim.x + threadIdx.x;
    int row = startRow + (idx >> 7);
    int o   = idx & 127;
    if (row >= N) return;
    const float* a = agg + (size_t)row * F_DIM;
    const float* w = W + o * F_DIM;
    float s = bias[o];
    for (int k = 0; k < F_DIM; ++k) s = fmaf(a[k], w[k], s);
    out[(size_t)row * F_DIM + o] = s;
}

// ------------------------------- host entry ---------------------------------
extern "C" void kernel_launch(void* const* d_in, const int* in_sizes, int n_in,
                              void* d_out, int out_size, void* d_ws, size_t ws_size,
                              hipStream_t stream) {
    const int*   L_rows = (const int*)d_in[0];
    const int*   L_cols = (const int*)d_in[1];
    const float* L_vals = (const float*)d_in[2];
    const float* x      = (const float*)d_in[3];
    const float* W      = (const float*)d_in[4];
    const float* b      = (const float*)d_in[5];
    float*       out    = (float*)d_out;
    float*       agg    = (float*)d_ws;          // N*128 fp32 scratch

    const int E = in_sizes[0];
    const int N = in_sizes[3] / F_DIM;

    // Phase 1: zero agg
    int n4 = (N * F_DIM) / 4;
    SCN_zero_kernel<<<(n4 + 255) / 256, 256, 0, stream>>>((float4*)agg, n4);

    // Phase 2: scatter (one wave32 per edge, 8 waves per block)
    SCN_scatter_kernel<<<(E + 7) / 8, 256, 0, stream>>>(L_rows, L_cols, L_vals,
                                                        x, agg, E);

    // Phase 3: WMMA GEMM (one wave per 16-node tile, 8 tiles per block)
    int nfull = N / 16;
    if (nfull > 0)
        SCN_gemm_wmma_kernel<<<(nfull + 7) / 8, 256, 0, stream>>>(agg, W, b, out,
                                                                  nfull);
    int rem = N & 15;
    if (rem > 0) {
        int nthr = rem * F_DIM;
        SCN_gemm_tail_kernel<<<(nthr + 127) / 128, 128, 0, stream>>>(
            agg, W, b, out, nfull * 16, N);
    }
}